// GNNCritic_82609400971716
// MI455X (gfx1250) — compile-verified
//
#include <hip/hip_runtime.h>

#define N_NODES 50000
#define N_EDGES 800000
#define N_GRAPHS 128
#define IN_DIM 12
#define C 64
#define NEG_SLOPE 0.01f

typedef __attribute__((ext_vector_type(2))) float v2f;
typedef __attribute__((ext_vector_type(8))) float v8f;

__device__ __forceinline__ float leaky(float v) {
    return v > 0.0f ? v : NEG_SLOPE * v;
}

// ---------------------------------------------------------------------------
// K1: read-in  h = lrelu([state|action] @ W_in + b_in)      [N, C]
// ---------------------------------------------------------------------------
__global__ void readin_kernel(const float* __restrict__ state,
                              const float* __restrict__ action,
                              const float* __restrict__ W_in,
                              const float* __restrict__ b_in,
                              float* __restrict__ h) {
    int idx = blockIdx.x * blockDim.x + threadIdx.x;
    if (idx >= N_NODES * C) return;
    int n = idx >> 6;
    int c = idx & 63;
    float acc = b_in[c];
#pragma unroll
    for (int k = 0; k < 8; ++k) acc += state[n * 8 + k] * W_in[k * C + c];
#pragma unroll
    for (int k = 0; k < 4; ++k) acc += action[n * 4 + k] * W_in[(8 + k) * C + c];
    h[idx] = leaky(acc);
}

// ---------------------------------------------------------------------------
// K2: aggE[dst] += edge_attr[e]   (layer-independent, computed once)
// ---------------------------------------------------------------------------
__global__ void edge_agg_attr(const int* __restrict__ ei,
                              const float* __restrict__ ea,
                              float* __restrict__ aggE) {
    long idx = (long)blockIdx.x * blockDim.x + threadIdx.x;
    if (idx >= (long)N_EDGES * IN_DIM) return;
    int e = (int)(idx / IN_DIM);
    int k = (int)(idx - (long)e * IN_DIM);
    int dst = ei[N_EDGES + e];
    atomicAdd(&aggE[dst * IN_DIM + k], ea[(long)e * IN_DIM + k]);
}

// ---------------------------------------------------------------------------
// K3: aggH[dst] += h[src]   (coalesced: 64 consecutive threads share an edge)
// ---------------------------------------------------------------------------
__global__ void edge_agg_h(const int* __restrict__ ei,
                           const float* __restrict__ h,
                           float* __restrict__ aggH) {
    int idx = blockIdx.x * blockDim.x + threadIdx.x;
    if (idx >= N_EDGES * C) return;
    int e = idx >> 6;
    int c = idx & 63;
    int src = ei[e];
    int dst = ei[N_EDGES + e];
    atomicAdd(&aggH[dst * C + c], h[src * C + c]);
}

// ---------------------------------------------------------------------------
// K4: node update via V_WMMA_F32_16X16X4_F32
//     hout = lrelu(hin@Wself + aggH@Wnbr + aggE@Wedge + bconv)
//     One wave handles a 16-node row block x all 64 output channels.
// ---------------------------------------------------------------------------
#define WMMA_F32(a, b, c) \
    __builtin_amdgcn_wmma_f32_16x16x4_f32(false, (a), false, (b), (short)0, (c), false, false)

__global__ void node_update_wmma(const float* __restrict__ hin,
                                 const float* __restrict__ aggH,
                                 const float* __restrict__ aggE,
                                 const float* __restrict__ Wself,
                                 const float* __restrict__ Wnbr,
                                 const float* __restrict__ Wedge,
                                 const float* __restrict__ bconv,
                                 float* __restrict__ hout) {
    int wave = blockIdx.x * (blockDim.x >> 5) + (threadIdx.x >> 5);
    int m0 = wave << 4;                 // 16 nodes per wave; wave-uniform guard
    if (m0 >= N_NODES) return;          // N_NODES % 16 == 0 -> EXEC all-ones

    int lane = threadIdx.x & 31;
    int half = lane >> 4;               // 0: K=k0,k0+1   1: K=k0+2,k0+3
    int lr = lane & 15;

    v8f acc0 = {}, acc1 = {}, acc2 = {}, acc3 = {};

    // ---- hin @ Wself  (K = 64) ----
    {
        const float* arow = hin + (m0 + lr) * C;
#pragma unroll
        for (int k0 = 0; k0 < C; k0 += 4) {
            int ka = k0 + 2 * half;
            v2f a; a.x = arow[ka]; a.y = arow[ka + 1];
            const float* br0 = Wself + ka * C;
            const float* br1 = Wself + (ka + 1) * C;
            v2f b0; b0.x = br0[lr];      b0.y = br1[lr];
            v2f b1; b1.x = br0[16 + lr]; b1.y = br1[16 + lr];
            v2f b2; b2.x = br0[32 + lr]; b2.y = br1[32 + lr];
            v2f b3; b3.x = br0[48 + lr]; b3.y = br1[48 + lr];
            acc0 = WMMA_F32(a, b0, acc0);
            acc1 = WMMA_F32(a, b1, acc1);
            acc2 = WMMA_F32(a, b2, acc2);
            acc3 = WMMA_F32(a, b3, acc3);
        }
    }
    // ---- aggH @ Wnbr  (K = 64) ----
    {
        const float* arow = aggH + (m0 + lr) * C;
#pragma unroll
        for (int k0 = 0; k0 < C; k0 += 4) {
            int ka = k0 + 2 * half;
            v2f a; a.x = arow[ka]; a.y = arow[ka + 1];
            const float* br0 = Wnbr + ka * C;
            const float* br1 = Wnbr + (ka + 1) * C;
            v2f b0; b0.x = br0[lr];      b0.y = br1[lr];
            v2f b1; b1.x = br0[16 + lr]; b1.y = br1[16 + lr];
            v2f b2; b2.x = br0[32 + lr]; b2.y = br1[32 + lr];
            v2f b3; b3.x = br0[48 + lr]; b3.y = br1[48 + lr];
            acc0 = WMMA_F32(a, b0, acc0);
            acc1 = WMMA_F32(a, b1, acc1);
            acc2 = WMMA_F32(a, b2, acc2);
            acc3 = WMMA_F32(a, b3, acc3);
        }
    }
    // ---- aggE @ Wedge  (K = 12) ----
    {
        const float* arow = aggE + (m0 + lr) * IN_DIM;
#pragma unroll
        for (int k0 = 0; k0 < IN_DIM; k0 += 4) {
            int ka = k0 + 2 * half;
            v2f a; a.x = arow[ka]; a.y = arow[ka + 1];
            const float* br0 = Wedge + ka * C;
            const float* br1 = Wedge + (ka + 1) * C;
            v2f b0; b0.x = br0[lr];      b0.y = br1[lr];
            v2f b1; b1.x = br0[16 + lr]; b1.y = br1[16 + lr];
            v2f b2; b2.x = br0[32 + lr]; b2.y = br1[32 + lr];
            v2f b3; b3.x = br0[48 + lr]; b3.y = br1[48 + lr];
            acc0 = WMMA_F32(a, b0, acc0);
            acc1 = WMMA_F32(a, b1, acc1);
            acc2 = WMMA_F32(a, b2, acc2);
            acc3 = WMMA_F32(a, b3, acc3);
        }
    }

    // ---- epilogue: bias + leaky relu + store ----
    // C/D layout: VGPR i -> row m0+i (lanes 0-15) / m0+8+i (lanes 16-31), col = n0+lr
#pragma unroll
    for (int t = 0; t < 4; ++t) {
        int n0 = t * 16;
        float bias = bconv[n0 + lr];
        v8f acc = (t == 0) ? acc0 : (t == 1) ? acc1 : (t == 2) ? acc2 : acc3;
#pragma unroll
        for (int i = 0; i < 8; ++i) {
            int row = m0 + i + 8 * half;
            float v = acc[i] + bias;
            hout[row * C + n0 + lr] = leaky(v);
        }
    }
}

// ---------------------------------------------------------------------------
// K5: readout  y = h @ W_out + b_out ; scatter sums/counts per graph
// ---------------------------------------------------------------------------
__global__ void readout_kernel(const float* __restrict__ h,
                               const float* __restrict__ W_out,
                               const float* __restrict__ b_out,
                               const int* __restrict__ batch,
                               float* __restrict__ sums,
                               float* __restrict__ counts) {
    int n = blockIdx.x * blockDim.x + threadIdx.x;
    if (n >= N_NODES) return;
    const float4* h4 = (const float4*)(h + (size_t)n * C);
    const float4* w4 = (const float4*)W_out;
    float acc = 0.0f;
#pragma unroll
    for (int k = 0; k < C / 4; ++k) {
        float4 a = h4[k];
        float4 w = w4[k];
        acc += a.x * w.x + a.y * w.y + a.z * w.z + a.w * w.w;
    }
    acc += b_out[0];
    int g = batch[n];
    atomicAdd(&sums[g], acc);
    atomicAdd(&counts[g], 1.0f);
}

// ---------------------------------------------------------------------------
// K6: finalize  out = sums / max(counts, 1)
// ---------------------------------------------------------------------------
__global__ void finalize_kernel(const float* __restrict__ sums,
                                const float* __restrict__ counts,
                                float* __restrict__ out) {
    int g = blockIdx.x * blockDim.x + threadIdx.x;
    if (g < N_GRAPHS) out[g] = sums[g] / fmaxf(counts[g], 1.0f);
}

// ---------------------------------------------------------------------------
extern "C" void kernel_launch(void* const* d_in, const int* in_sizes, int n_in,
                              void* d_out, int out_size, void* d_ws, size_t ws_size,
                              hipStream_t stream) {
    const float* state  = (const float*)d_in[0];
    const float* action = (const float*)d_in[1];
    const int*   ei     = (const int*)d_in[2];    // [2, E]
    const float* ea     = (const float*)d_in[3];  // [E, 12]
    const int*   batch  = (const int*)d_in[4];    // [N]
    const float* W_in   = (const float*)d_in[5];  // [12, 64]
    const float* b_in   = (const float*)d_in[6];  // [64]
    const float* W_self = (const float*)d_in[7];  // [2, 64, 64]
    const float* W_nbr  = (const float*)d_in[8];  // [2, 64, 64]
    const float* W_edge = (const float*)d_in[9];  // [2, 12, 64]
    const float* b_conv = (const float*)d_in[10]; // [2, 64]
    const float* W_out  = (const float*)d_in[11]; // [64, 1]
    const float* b_out  = (const float*)d_in[12]; // [1]
    float* out = (float*)d_out;

    float* ws     = (float*)d_ws;
    float* h      = ws;                                   // N*C
    float* h2     = h    + (size_t)N_NODES * C;           // N*C
    float* aggH   = h2   + (size_t)N_NODES * C;           // N*C
    float* aggE   = aggH + (size_t)N_NODES * C;           // N*12
    float* sums   = aggE + (size_t)N_NODES * IN_DIM;      // 128
    float* counts = sums + N_GRAPHS;                      // 128

    // zero accumulators (graph-capturable memset nodes)
    hipMemsetAsync(aggE, 0, sizeof(float) * (size_t)N_NODES * IN_DIM, stream);
    hipMemsetAsync(sums, 0, sizeof(float) * 2 * N_GRAPHS, stream);

    // read-in
    {
        int tot = N_NODES * C;
        readin_kernel<<<(tot + 255) / 256, 256, 0, stream>>>(state, action, W_in, b_in, h);
    }
    // edge-attr aggregation (layer-independent)
    {
        long tot = (long)N_EDGES * IN_DIM;
        edge_agg_attr<<<(int)((tot + 255) / 256), 256, 0, stream>>>(ei, ea, aggE);
    }

    const float* hin = h;
    float* hout = h2;
    for (int l = 0; l < 2; ++l) {
        hipMemsetAsync(aggH, 0, sizeof(float) * (size_t)N_NODES * C, stream);
        int totEH = N_EDGES * C;
        edge_agg_h<<<(totEH + 255) / 256, 256, 0, stream>>>(ei, hin, aggH);

        int nwaves = N_NODES / 16;              // 3125 row blocks
        int blocks = (nwaves + 7) / 8;          // 8 waves / block (256 threads)
        node_update_wmma<<<blocks, 256, 0, stream>>>(
            hin, aggH, aggE,
            W_self + (size_t)l * C * C,
            W_nbr  + (size_t)l * C * C,
            W_edge + (size_t)l * IN_DIM * C,
            b_conv + (size_t)l * C,
            hout);

        const float* t = hin; hin = hout; hout = (float*)t;
    }

    readout_kernel<<<(N_NODES + 255) / 256, 256, 0, stream>>>(hin, W_out, b_out,
                                                              batch, sums, counts);
    finalize_kernel<<<1, N_GRAPHS, 0, stream>>>(sums, counts, out);
}